// MoNetDiscriminator_52999896432810
// MI455X (gfx1250) — compile-verified
//
#include <hip/hip_runtime.h>
#include <math.h>

// ---------------------------------------------------------------------------
// MoNet discriminator pipeline for gfx1250 (MI455X), wave32.
// Dense GEMMs (xp = x@g, root GEMMs, fc1) use V_WMMA_F32_16X16X4_F32.
// Edge message passing is wave-per-edge (25 Gaussian weights broadcast via
// __shfl within the 32-lane wave), scatter via global atomics.
// ---------------------------------------------------------------------------

typedef float v2f __attribute__((ext_vector_type(2)));
typedef float v8f __attribute__((ext_vector_type(8)));

#define KGM 25
#define EPSK 1e-15f

static constexpr int N0 = 131072, E1 = 1048576;
static constexpr int N1 = 65536,  E2 = 524288;
static constexpr int N2 = 32768,  E3 = 262144;
static constexpr int NB = 128;

// workspace layout (float offsets)
static constexpr size_t OFF_XP   = 0;                       // 65536*1600
static constexpr size_t OFF_S    = OFF_XP   + 104857600;    // max N0*32 / N1*64
static constexpr size_t OFF_CNT  = OFF_S    + 4194304;      // N0
static constexpr size_t OFF_X1   = OFF_CNT  + 131072;       // N0*32
static constexpr size_t OFF_XM1  = OFF_X1   + 4194304;      // N1*32
static constexpr size_t OFF_POS1 = OFF_XM1  + 2097152;      // N1*2
static constexpr size_t OFF_PCNT = OFF_POS1 + 131072;       // N1
static constexpr size_t OFF_XR   = OFF_PCNT + 131072;       // N1*64
static constexpr size_t OFF_X2   = OFF_XR   + 4194304;      // N1*64
static constexpr size_t OFF_XM2  = OFF_X2   + 4194304;      // N2*64
static constexpr size_t OFF_POS2 = OFF_XM2  + 2097152;      // N2*2
static constexpr size_t OFF_X3   = OFF_POS2 + 65536;        // N2*64
static constexpr size_t OFF_H    = OFF_X3   + 2097152;      // N2*128
static constexpr size_t OFF_BSUM = OFF_H    + 4194304;      // 128
static constexpr size_t OFF_BCNT = OFF_BSUM + 128;          // 128
static constexpr size_t OFF_MAX  = OFF_BCNT + 128;          // 2 scalars

__device__ __forceinline__ float eluf(float v) {
  return v > 0.0f ? v : (__expf(v) - 1.0f);
}

__device__ __forceinline__ void atomicMaxFloat(float* addr, float val) {
  // sign-split trick: positives ordered as int, negatives reverse-ordered as uint
  if (val >= 0.0f) atomicMax((int*)addr, __float_as_int(val));
  else             atomicMin((unsigned int*)addr, __float_as_uint(val));
}

// ---------------------------------------------------------------------------
__global__ void fill_f32(float* p, float v, int n) {
  int i = blockIdx.x * blockDim.x + threadIdx.x;
  int stride = gridDim.x * blockDim.x;
  for (; i < n; i += stride) p[i] = v;
}

// ---------------------------------------------------------------------------
// C[M,N] = act(A[M,Kd] @ B[Kd,N] + bias)  via v_wmma_f32_16x16x4_f32
// One 16x16 output tile per wave; 4 waves / 128-thread block.
// A fragment (16x4 f32): lanes 0-15 hold M rows, v0=K0/v1=K1; lanes 16-31 K2/K3.
// B fragment (4x16): v0 = rows K0(K2), v1 = rows K1(K3), N striped over lanes.
// C/D: vgpr r -> (M = r + 8*hi, N = lane&15).
__global__ void gemm_wmma_f32(const float* __restrict__ A,
                              const float* __restrict__ B,
                              float* __restrict__ C,
                              int M, int N, int Kd,
                              const float* __restrict__ bias, int act) {
  const int wave = threadIdx.x >> 5;
  const int lane = threadIdx.x & 31;
  const int nTiles = N >> 4;
  const int tn = blockIdx.x * 4 + wave;
  const int tm = blockIdx.y;
  if (tn >= nTiles) return;                 // uniform per wave: EXEC all-ones inside
  const int l15 = lane & 15;
  const int hi  = lane >> 4;
  const int arow = tm * 16 + l15;
  const int bcol = tn * 16 + l15;
  v8f c = {0.f, 0.f, 0.f, 0.f, 0.f, 0.f, 0.f, 0.f};
  for (int k0 = 0; k0 < Kd; k0 += 4) {
    const int ka = k0 + (hi << 1);
    v2f a, b;
    a.x = A[arow * Kd + ka];
    a.y = A[arow * Kd + ka + 1];
    b.x = B[ka * N + bcol];
    b.y = B[(ka + 1) * N + bcol];
    c = __builtin_amdgcn_wmma_f32_16x16x4_f32(
        /*neg_a=*/false, a, /*neg_b=*/false, b,
        /*c_mod=*/(short)0, c, /*reuse_a=*/false, /*reuse_b=*/false);
  }
  const float bv = bias ? bias[bcol] : 0.0f;
#pragma unroll
  for (int r = 0; r < 8; ++r) {
    const int m = tm * 16 + r + (hi << 3);
    float v = c[r] + bv;
    if (act) v = eluf(v);
    C[m * N + bcol] = v;
  }
}

// ---------------------------------------------------------------------------
// Layer 1 edge messages (in=1, out=32): msg_e = x[row] * (w_e @ g1[K,32])
__global__ void msg1_kernel(const float* __restrict__ x,
                            const float* __restrict__ pseudo,
                            const int* __restrict__ ei, int E,
                            const float* __restrict__ g1,
                            const float* __restrict__ mu,
                            const float* __restrict__ sigma,
                            float* __restrict__ S, float* __restrict__ CNT) {
  const int wid  = (blockIdx.x * blockDim.x + threadIdx.x) >> 5;
  const int lane = threadIdx.x & 31;
  if (wid >= E) return;
  const int row = ei[wid];
  const int col = ei[E + wid];
  const float p0 = pseudo[wid * 2], p1 = pseudo[wid * 2 + 1];
  float wl = 0.0f;
  if (lane < KGM) {
    const float d0 = p0 - mu[lane * 2],     d1 = p1 - mu[lane * 2 + 1];
    const float s0 = sigma[lane * 2],       s1 = sigma[lane * 2 + 1];
    wl = __expf(-0.5f * (d0 * d0 / (EPSK + s0 * s0) + d1 * d1 / (EPSK + s1 * s1)));
  }
  float acc = 0.0f;
#pragma unroll
  for (int k = 0; k < KGM; ++k) {
    const float wk = __shfl(wl, k, 32);
    acc += wk * g1[k * 32 + lane];
  }
  atomicAdd(&S[(size_t)col * 32 + lane], x[row] * acc);
  if (lane == 0) atomicAdd(&CNT[col], 1.0f);
}

// Layers 2/3 edge messages (out=64): pseudo from Cartesian(pos), xp gathered.
__global__ void msg64_kernel(const float* __restrict__ xp,
                             const float* __restrict__ pos,
                             const int* __restrict__ ei, int E,
                             const float* __restrict__ mu,
                             const float* __restrict__ sigma,
                             const float* __restrict__ maxabs,
                             float* __restrict__ S, float* __restrict__ CNT) {
  const int wid  = (blockIdx.x * blockDim.x + threadIdx.x) >> 5;
  const int lane = threadIdx.x & 31;
  if (wid >= E) return;
  const int row = ei[wid];
  const int col = ei[E + wid];
  const float inv = 0.5f / fmaxf(maxabs[0], 1e-30f);
  const float p0 = (pos[col * 2]     - pos[row * 2])     * inv + 0.5f;
  const float p1 = (pos[col * 2 + 1] - pos[row * 2 + 1]) * inv + 0.5f;
  float wl = 0.0f;
  if (lane < KGM) {
    const float d0 = p0 - mu[lane * 2],     d1 = p1 - mu[lane * 2 + 1];
    const float s0 = sigma[lane * 2],       s1 = sigma[lane * 2 + 1];
    wl = __expf(-0.5f * (d0 * d0 / (EPSK + s0 * s0) + d1 * d1 / (EPSK + s1 * s1)));
  }
  const float* xr = xp + (size_t)row * (KGM * 64);
  float a0 = 0.0f, a1 = 0.0f;
#pragma unroll
  for (int k = 0; k < KGM; ++k) {
    const float wk = __shfl(wl, k, 32);
    a0 += wk * xr[k * 64 + lane];
    a1 += wk * xr[k * 64 + 32 + lane];
  }
  atomicAdd(&S[(size_t)col * 64 + lane],      a0);
  atomicAdd(&S[(size_t)col * 64 + 32 + lane], a1);
  if (lane == 0) atomicAdd(&CNT[col], 1.0f);
}

// ---------------------------------------------------------------------------
__global__ void combine1_kernel(const float* __restrict__ S, const float* __restrict__ CNT,
                                const float* __restrict__ x,
                                const float* __restrict__ root1, const float* __restrict__ b1,
                                float* __restrict__ X1, int n) {
  const int i = blockIdx.x * blockDim.x + threadIdx.x;
  if (i >= n) return;
  const int nd = i >> 5, o = i & 31;
  const float cnt = fmaxf(CNT[nd], 1.0f);
  X1[i] = eluf(S[i] / cnt + x[nd] * root1[o] + b1[o]);
}

__global__ void combine64_kernel(const float* __restrict__ S, const float* __restrict__ CNT,
                                 const float* __restrict__ XR, const float* __restrict__ b,
                                 float* __restrict__ X, int n) {
  const int i = blockIdx.x * blockDim.x + threadIdx.x;
  if (i >= n) return;
  const int nd = i >> 6, o = i & 63;
  const float cnt = fmaxf(CNT[nd], 1.0f);
  X[i] = eluf(S[i] / cnt + XR[i] + b[o]);
}

// ---------------------------------------------------------------------------
__global__ void pool_max_kernel(const float* __restrict__ X, const int* __restrict__ cluster,
                                int n, int F, float* __restrict__ XM) {
  const int i = blockIdx.x * blockDim.x + threadIdx.x;
  if (i >= n * F) return;
  const int nd = i / F, o = i % F;
  atomicMaxFloat(&XM[(size_t)cluster[nd] * F + o], X[i]);
}

__global__ void xm_fix_kernel(float* __restrict__ XM, int n) {
  const int i = blockIdx.x * blockDim.x + threadIdx.x;
  if (i >= n) return;
  const float v = XM[i];
  if (!isfinite(v)) XM[i] = 0.0f;      // empty clusters -> 0 (reference semantics)
}

__global__ void pos_accum_kernel(const float* __restrict__ pos, const int* __restrict__ cluster,
                                 int n, float* __restrict__ PSUM, float* __restrict__ PCNT) {
  const int i = blockIdx.x * blockDim.x + threadIdx.x;
  if (i >= n) return;
  const int c = cluster[i];
  atomicAdd(&PSUM[c * 2],     pos[i * 2]);
  atomicAdd(&PSUM[c * 2 + 1], pos[i * 2 + 1]);
  atomicAdd(&PCNT[c], 1.0f);
}

__global__ void pos_div_kernel(float* __restrict__ PSUM, const float* __restrict__ PCNT, int nc) {
  const int i = blockIdx.x * blockDim.x + threadIdx.x;
  if (i >= nc) return;
  const float inv = 1.0f / fmaxf(PCNT[i], 1.0f);
  PSUM[i * 2]     *= inv;
  PSUM[i * 2 + 1] *= inv;
}

// ---------------------------------------------------------------------------
__global__ void cart_max_kernel(const float* __restrict__ pos, const int* __restrict__ ei,
                                int E, float* __restrict__ maxabs) {
  const int i = blockIdx.x * blockDim.x + threadIdx.x;
  float m = 0.0f;
  if (i < E) {
    const int r = ei[i], c = ei[E + i];
    m = fmaxf(fabsf(pos[c * 2] - pos[r * 2]), fabsf(pos[c * 2 + 1] - pos[r * 2 + 1]));
  }
#pragma unroll
  for (int off = 16; off; off >>= 1) m = fmaxf(m, __shfl_xor(m, off, 32));
  if ((threadIdx.x & 31) == 0) atomicMax((int*)maxabs, __float_as_int(m));
}

// ---------------------------------------------------------------------------
// fc2 + global mean-pool accumulation: one wave per node.
__global__ void head_kernel(const float* __restrict__ H, const float* __restrict__ w2,
                            const float* __restrict__ b2, const int* __restrict__ batch,
                            int n, float* __restrict__ BSUM, float* __restrict__ BCNT) {
  const int wid  = (blockIdx.x * blockDim.x + threadIdx.x) >> 5;
  const int lane = threadIdx.x & 31;
  if (wid >= n) return;
  const float* h = H + (size_t)wid * 128;
  float p = 0.0f;
#pragma unroll
  for (int j = lane; j < 128; j += 32) p += h[j] * w2[j];
#pragma unroll
  for (int off = 16; off; off >>= 1) p += __shfl_down(p, off, 32);
  if (lane == 0) {
    const int b = batch[wid];
    atomicAdd(&BSUM[b], p + b2[0]);
    atomicAdd(&BCNT[b], 1.0f);
  }
}

__global__ void final_kernel(const float* __restrict__ BSUM, const float* __restrict__ BCNT,
                             float* __restrict__ out, int nb) {
  const int i = blockIdx.x * blockDim.x + threadIdx.x;
  if (i >= nb) return;
  const float y = BSUM[i] / fmaxf(BCNT[i], 1.0f);
  out[i] = 1.0f / (1.0f + expf(-y));
}

// ---------------------------------------------------------------------------
extern "C" void kernel_launch(void* const* d_in, const int* in_sizes, int n_in,
                              void* d_out, int out_size, void* d_ws, size_t ws_size,
                              hipStream_t stream) {
  const float* x      = (const float*)d_in[0];
  const float* pos    = (const float*)d_in[1];
  const float* ea1    = (const float*)d_in[2];
  const int*   ei1    = (const int*)d_in[3];
  const int*   cl1    = (const int*)d_in[4];
  const int*   ei2    = (const int*)d_in[5];
  const int*   cl2    = (const int*)d_in[6];
  const int*   ei3    = (const int*)d_in[7];
  const int*   batch3 = (const int*)d_in[8];
  const float* g1     = (const float*)d_in[9];
  const float* mu1    = (const float*)d_in[10];
  const float* sg1    = (const float*)d_in[11];
  const float* root1  = (const float*)d_in[12];
  const float* b1     = (const float*)d_in[13];
  const float* g2     = (const float*)d_in[14];
  const float* mu2    = (const float*)d_in[15];
  const float* sg2    = (const float*)d_in[16];
  const float* root2  = (const float*)d_in[17];
  const float* b2     = (const float*)d_in[18];
  const float* g3     = (const float*)d_in[19];
  const float* mu3    = (const float*)d_in[20];
  const float* sg3    = (const float*)d_in[21];
  const float* root3  = (const float*)d_in[22];
  const float* b3     = (const float*)d_in[23];
  const float* fc1w   = (const float*)d_in[24];
  const float* fc1b   = (const float*)d_in[25];
  const float* fc2w   = (const float*)d_in[26];
  const float* fc2b   = (const float*)d_in[27];
  float* out = (float*)d_out;

  float* W = (float*)d_ws;
  float* XP   = W + OFF_XP;
  float* S    = W + OFF_S;
  float* CNT  = W + OFF_CNT;
  float* X1   = W + OFF_X1;
  float* XM1  = W + OFF_XM1;
  float* POS1 = W + OFF_POS1;
  float* PCNT = W + OFF_PCNT;
  float* XR   = W + OFF_XR;
  float* X2   = W + OFF_X2;
  float* XM2  = W + OFF_XM2;
  float* POS2 = W + OFF_POS2;
  float* X3   = W + OFF_X3;
  float* H    = W + OFF_H;
  float* BSUM = W + OFF_BSUM;
  float* BCNT = W + OFF_BCNT;
  float* MAXA = W + OFF_MAX;

  const int T = 256;
  auto blocks = [](long n, int t) { return (int)((n + t - 1) / t); };

  // ---- Layer 1: GMMConv(1 -> 32) on (x, edge_attr1), N0 nodes, E1 edges ----
  fill_f32<<<2048, T, 0, stream>>>(S, 0.0f, N0 * 32);
  fill_f32<<<512,  T, 0, stream>>>(CNT, 0.0f, N0);
  msg1_kernel<<<blocks((long)E1 * 32, T), T, 0, stream>>>(x, ea1, ei1, E1, g1, mu1, sg1, S, CNT);
  combine1_kernel<<<blocks((long)N0 * 32, T), T, 0, stream>>>(S, CNT, x, root1, b1, X1, N0 * 32);

  // ---- Pool 1: cluster1 (N0 -> N1), features max, positions mean ----
  fill_f32<<<2048, T, 0, stream>>>(XM1, -INFINITY, N1 * 32);
  fill_f32<<<512,  T, 0, stream>>>(POS1, 0.0f, N1 * 2);
  fill_f32<<<512,  T, 0, stream>>>(PCNT, 0.0f, N1);
  pool_max_kernel<<<blocks((long)N0 * 32, T), T, 0, stream>>>(X1, cl1, N0, 32, XM1);
  pos_accum_kernel<<<blocks(N0, T), T, 0, stream>>>(pos, cl1, N0, POS1, PCNT);
  xm_fix_kernel<<<blocks((long)N1 * 32, T), T, 0, stream>>>(XM1, N1 * 32);
  pos_div_kernel<<<blocks(N1, T), T, 0, stream>>>(POS1, PCNT, N1);

  // ---- Layer 2: Cartesian pseudo + GMMConv(32 -> 64) on N1, E2 ----
  fill_f32<<<1, 64, 0, stream>>>(MAXA, 0.0f, 2);
  cart_max_kernel<<<blocks(E2, T), T, 0, stream>>>(POS1, ei2, E2, MAXA);
  {  // XP = XM1[65536,32] @ g2[32,1600]  (WMMA), XR = XM1 @ root2[32,64]
    dim3 g((100 + 3) / 4, N1 / 16);
    gemm_wmma_f32<<<g, 128, 0, stream>>>(XM1, g2, XP, N1, 1600, 32, nullptr, 0);
    dim3 gr(1, N1 / 16);
    gemm_wmma_f32<<<gr, 128, 0, stream>>>(XM1, root2, XR, N1, 64, 32, nullptr, 0);
  }
  fill_f32<<<2048, T, 0, stream>>>(S, 0.0f, N1 * 64);
  fill_f32<<<512,  T, 0, stream>>>(CNT, 0.0f, N1);
  msg64_kernel<<<blocks((long)E2 * 32, T), T, 0, stream>>>(XP, POS1, ei2, E2, mu2, sg2, MAXA, S, CNT);
  combine64_kernel<<<blocks((long)N1 * 64, T), T, 0, stream>>>(S, CNT, XR, b2, X2, N1 * 64);

  // ---- Pool 2: cluster2 (N1 -> N2) ----
  fill_f32<<<2048, T, 0, stream>>>(XM2, -INFINITY, N2 * 64);
  fill_f32<<<512,  T, 0, stream>>>(POS2, 0.0f, N2 * 2);
  fill_f32<<<512,  T, 0, stream>>>(PCNT, 0.0f, N2);
  pool_max_kernel<<<blocks((long)N1 * 64, T), T, 0, stream>>>(X2, cl2, N1, 64, XM2);
  pos_accum_kernel<<<blocks(N1, T), T, 0, stream>>>(POS1, cl2, N1, POS2, PCNT);
  xm_fix_kernel<<<blocks((long)N2 * 64, T), T, 0, stream>>>(XM2, N2 * 64);
  pos_div_kernel<<<blocks(N2, T), T, 0, stream>>>(POS2, PCNT, N2);

  // ---- Layer 3: Cartesian pseudo + GMMConv(64 -> 64) on N2, E3 ----
  fill_f32<<<1, 64, 0, stream>>>(MAXA + 1, 0.0f, 1);
  cart_max_kernel<<<blocks(E3, T), T, 0, stream>>>(POS2, ei3, E3, MAXA + 1);
  {  // XP = XM2[32768,64] @ g3[64,1600], XR = XM2 @ root3[64,64]
    dim3 g((100 + 3) / 4, N2 / 16);
    gemm_wmma_f32<<<g, 128, 0, stream>>>(XM2, g3, XP, N2, 1600, 64, nullptr, 0);
    dim3 gr(1, N2 / 16);
    gemm_wmma_f32<<<gr, 128, 0, stream>>>(XM2, root3, XR, N2, 64, 64, nullptr, 0);
  }
  fill_f32<<<2048, T, 0, stream>>>(S, 0.0f, N2 * 64);
  fill_f32<<<512,  T, 0, stream>>>(CNT, 0.0f, N2);
  msg64_kernel<<<blocks((long)E3 * 32, T), T, 0, stream>>>(XP, POS2, ei3, E3, mu3, sg3, MAXA + 1, S, CNT);
  combine64_kernel<<<blocks((long)N2 * 64, T), T, 0, stream>>>(S, CNT, XR, b3, X3, N2 * 64);

  // ---- Head: fc1 (WMMA, fused bias+ELU), fc2 + global mean pool, sigmoid ----
  {
    dim3 g((8 + 3) / 4, N2 / 16);  // N=128 -> 8 tiles
    gemm_wmma_f32<<<g, 128, 0, stream>>>(X3, fc1w, H, N2, 128, 64, fc1b, 1);
  }
  fill_f32<<<1, 256, 0, stream>>>(BSUM, 0.0f, NB);
  fill_f32<<<1, 256, 0, stream>>>(BCNT, 0.0f, NB);
  head_kernel<<<blocks((long)N2 * 32, T), T, 0, stream>>>(H, fc2w, fc2b, batch3, N2, BSUM, BCNT);
  final_kernel<<<1, NB, 0, stream>>>(BSUM, BCNT, out, NB);
}